// FlashAttention_55396488183810
// MI455X (gfx1250) — compile-verified
//
#include <hip/hip_runtime.h>

typedef float v8f __attribute__((ext_vector_type(8)));
typedef float v4f __attribute__((ext_vector_type(4)));
typedef float v2f __attribute__((ext_vector_type(2)));

constexpr int   B_      = 4;
constexpr int   H_      = 32;
constexpr int   D_      = 128;
constexpr int   S_      = 8192;
constexpr int   NSPLIT  = 4;
constexpr int   CHUNK   = S_ / NSPLIT;   // 2048 positions per split
constexpr int   WAVES   = 8;             // waves per block (wave32)
constexpr int   TILE    = 16;            // positions per WMMA tile
constexpr float SCALE   = 0.08838834764831845f; // 1/sqrt(128)
constexpr float NEGINF  = -1e30f;

// LDS K-tile staging: row stride 132 dwords (528B) -> banks spread 4/row,
// ds_load_b64 fragment gather is conflict-free across all 32 lanes.
constexpr int ROWF   = 132;              // floats per padded LDS row
constexpr int TILEF  = TILE * ROWF;      // floats per K tile buffer (2112)
constexpr int KSTAGE = 2 * TILEF;        // double buffer per wave (4224)
constexpr int SMEM_F = WAVES * KSTAGE + WAVES + WAVES + WAVES * D_; // 34832

// ---------------------------------------------------------------------------
// Kernel 1: per-(b,h,split) flash-decoding partial.
//   K tiles: async DMA to LDS (double buffered, ASYNCcnt) -> ds_load_b64
//            A-fragments -> v_wmma_f32_16x16x4_f32
//   V rows : direct coalesced b128 non-temporal loads
// ---------------------------------------------------------------------------
__global__ __launch_bounds__(WAVES * 32)
void attn_partial(const float* __restrict__ q,    // [B,1,H,D]
                  const float* __restrict__ kv,   // [2,B,S,H,D]
                  const int*   __restrict__ posp, // scalar current_pos
                  float* __restrict__ wsM,        // [B*H*NSPLIT]
                  float* __restrict__ wsL,        // [B*H*NSPLIT]
                  float* __restrict__ wsAcc)      // [B*H*NSPLIT, D]
{
    extern __shared__ float smem[];

    const int blk   = blockIdx.x;
    const int split = blk % NSPLIT;
    const int bh    = blk / NSPLIT;
    const int b     = bh / H_;
    const int h     = bh % H_;
    const int tid   = threadIdx.x;
    const int wave  = tid >> 5;
    const int lane  = tid & 31;
    const int half  = lane >> 4;   // 0: rows 0-7 / K 0-1 ; 1: rows 8-15 / K 2-3
    const int l16   = lane & 15;

    const int svalid = *posp + 1;  // attend to s <= current_pos

    const size_t rowStride = (size_t)H_ * D_;
    const float* qp = q  + (size_t)bh * D_;
    const float* Kb = kv + (size_t)b * S_ * rowStride + (size_t)h * D_;
    const float* Vb = Kb + (size_t)B_ * S_ * rowStride;

    float* kstage = smem + wave * KSTAGE;            // this wave's K staging
    float* lm     = smem + WAVES * KSTAGE;           // [WAVES]
    float* ll     = lm + WAVES;                      // [WAVES]
    float* lacc   = ll + WAVES;                      // [WAVES][D]

    // q fragment for WMMA B-matrix (4x16, columns all equal):
    // per K-chunk i (d0 = 4*i): lanes 0-15 hold K=0,1 ; lanes 16-31 hold K=2,3
    v2f qf[32];
#pragma unroll
    for (int i = 0; i < 32; ++i) {
        qf[i].x = qp[4 * i + 2 * half + 0];
        qf[i].y = qp[4 * i + 2 * half + 1];
    }

    // async-DMA one 16xD K tile (s0 = first row) into staging buffer `buf`
    auto stageK = [&](int s0, int buf) {
        const float* gsrc = Kb + (size_t)s0 * rowStride + 4 * lane;  // 16B/lane
        float*       ldst = kstage + buf * TILEF + 4 * lane;
#pragma unroll
        for (int r = 0; r < 16; ++r) {
            unsigned     loff = (unsigned)(uintptr_t)(ldst + r * ROWF);
            const float* gp   = gsrc + (size_t)r * rowStride;
            asm volatile("global_load_async_to_lds_b128 %0, %1, off th:TH_LOAD_NT"
                         :: "v"(loff), "v"(gp) : "memory");
        }
    };

    float m_run = NEGINF;
    float l_run = 0.f;
    v4f   acc   = {0.f, 0.f, 0.f, 0.f};

    const int base   = split * CHUNK;
    const int ntiles = CHUNK / TILE;
    int bufc = 0;

    if (wave < ntiles && base + wave * TILE < svalid)   // prefetch first tile
        stageK(base + wave * TILE, 0);

    for (int t = wave; t < ntiles; t += WAVES) {
        const int s0 = base + t * TILE;
        if (s0 >= svalid) break;   // wave-uniform: EXEC stays all-ones

        // prefetch next tile into other buffer, then wait for current buffer
        const int  tn = t + WAVES;
        const bool pn = (tn < ntiles) && (base + tn * TILE < svalid);
        if (pn) {
            stageK(base + tn * TILE, bufc ^ 1);
            asm volatile("s_wait_asynccnt 0x10" ::: "memory"); // current 16 done
        } else {
            asm volatile("s_wait_asynccnt 0x0"  ::: "memory");
        }

        // ---- scores[16] = K_tile(16xD) . q  via 32x v_wmma_f32_16x16x4_f32
        v8f c = {};
        const float* kb = kstage + bufc * TILEF + l16 * ROWF + 2 * half;
#pragma unroll
        for (int i = 0; i < 32; ++i) {
            v2f a = *(const v2f*)(kb + 4 * i);   // ds_load_b64, conflict-free
            c = __builtin_amdgcn_wmma_f32_16x16x4_f32(
                    false, a, false, qf[i], (short)0, c, false, false);
        }
        bufc ^= 1;

        // C layout: VGPR r -> row (r + 8*half); all 16 columns identical.
        float p[8];
        float tmax = NEGINF;
#pragma unroll
        for (int r = 0; r < 8; ++r) {
            const int srow = s0 + r + 8 * half;
            const float sc = (srow < svalid) ? c[r] * SCALE : NEGINF;
            p[r] = sc;
            tmax = fmaxf(tmax, sc);
        }
        tmax = fmaxf(tmax, __shfl_xor(tmax, 16, 32));  // merge the two halves

        const float m_new = fmaxf(m_run, tmax);
        const float corr  = __expf(m_run - m_new);
        float lsum = 0.f;
#pragma unroll
        for (int r = 0; r < 8; ++r) {
            p[r] = __expf(p[r] - m_new);
            lsum += p[r];
        }
        l_run = l_run * corr + lsum + __shfl_xor(lsum, 16, 32);
        m_run = m_new;
        acc.x *= corr; acc.y *= corr; acc.z *= corr; acc.w *= corr;

        // p for the other half's 8 rows
        float po[8];
#pragma unroll
        for (int r = 0; r < 8; ++r) po[r] = __shfl_xor(p[r], 16, 32);

        // ---- PV: lane owns d = 4*lane .. 4*lane+3 (coalesced 512B rows)
        const float* vrow = Vb + (size_t)s0 * rowStride + 4 * lane;
#pragma unroll
        for (int i = 0; i < 16; ++i) {
            const float pr = ((i >> 3) == half) ? p[i & 7] : po[i & 7];
            v4f v = __builtin_nontemporal_load((const v4f*)(vrow + (size_t)i * rowStride));
            acc.x += pr * v.x; acc.y += pr * v.y;
            acc.z += pr * v.z; acc.w += pr * v.w;
        }
    }

    // ---- merge the 8 waves through LDS -----------------------------------
    lacc[wave * D_ + 4 * lane + 0] = acc.x;
    lacc[wave * D_ + 4 * lane + 1] = acc.y;
    lacc[wave * D_ + 4 * lane + 2] = acc.z;
    lacc[wave * D_ + 4 * lane + 3] = acc.w;
    if (lane == 0) { lm[wave] = m_run; ll[wave] = l_run; }
    __syncthreads();

    if (tid < D_) {
        float M = NEGINF;
#pragma unroll
        for (int w = 0; w < WAVES; ++w) M = fmaxf(M, lm[w]);
        float L = 0.f, a = 0.f;
#pragma unroll
        for (int w = 0; w < WAVES; ++w) {
            const float f = __expf(lm[w] - M);
            L += ll[w] * f;
            a += lacc[w * D_ + tid] * f;
        }
        const int g = bh * NSPLIT + split;
        wsAcc[(size_t)g * D_ + tid] = a;
        if (tid == 0) { wsM[g] = M; wsL[g] = L; }
    }
}

// ---------------------------------------------------------------------------
// Kernel 2: combine the NSPLIT partials per (b,h)
// ---------------------------------------------------------------------------
__global__ __launch_bounds__(D_)
void attn_merge(const float* __restrict__ wsM,
                const float* __restrict__ wsL,
                const float* __restrict__ wsAcc,
                float* __restrict__ out)           // [B,1,H,D]
{
    const int bh = blockIdx.x;
    const int d  = threadIdx.x;

    float M = NEGINF;
#pragma unroll
    for (int s = 0; s < NSPLIT; ++s) M = fmaxf(M, wsM[bh * NSPLIT + s]);

    float L = 0.f, a = 0.f;
#pragma unroll
    for (int s = 0; s < NSPLIT; ++s) {
        const int   g = bh * NSPLIT + s;
        const float f = __expf(wsM[g] - M);
        L += wsL[g] * f;
        a += wsAcc[(size_t)g * D_ + d] * f;
    }
    out[(size_t)bh * D_ + d] = a / L;
}

// ---------------------------------------------------------------------------
extern "C" void kernel_launch(void* const* d_in, const int* in_sizes, int n_in,
                              void* d_out, int out_size, void* d_ws, size_t ws_size,
                              hipStream_t stream)
{
    (void)in_sizes; (void)n_in; (void)out_size; (void)ws_size;

    const float* q   = (const float*)d_in[0];
    const float* kv  = (const float*)d_in[1];
    const int*   pos = (const int*)d_in[2];
    float*       out = (float*)d_out;

    const int G = B_ * H_ * NSPLIT;           // 512 partials
    float* wsM   = (float*)d_ws;
    float* wsL   = wsM + G;
    float* wsAcc = wsL + G;                   // G*128 floats (~256 KB total)

    const size_t shmem = (size_t)SMEM_F * sizeof(float);  // ~139 KB
    attn_partial<<<dim3(G), dim3(WAVES * 32), shmem, stream>>>(q, kv, pos, wsM, wsL, wsAcc);
    attn_merge<<<dim3(B_ * H_), dim3(D_), 0, stream>>>(wsM, wsL, wsAcc, out);
}